// ConvAttentionLayer_11708080849271
// MI455X (gfx1250) — compile-verified
//
#include <hip/hip_runtime.h>

typedef float v2f __attribute__((ext_vector_type(2)));
typedef float v8f __attribute__((ext_vector_type(8)));

#define CC 64
#define HH 256
#define WW 256
#define HWSZ (HH * WW)
#define CHW (CC * HWSZ)

// One wave handles a tile of 16 consecutive pixels along W (fixed b, h).
// Block = 256 threads = 8 waves = 128 pixels of one row.
__global__ __launch_bounds__(256) void conv_attn_kernel(
    const float* __restrict__ q, const float* __restrict__ k,
    const float* __restrict__ v, float* __restrict__ out)
{
    __shared__ float sS[8][256];   // per-wave 16x16 score tile

    const int tid  = threadIdx.x;
    const int wave = tid >> 5;
    const int lane = tid & 31;
    const int p    = lane & 15;    // pixel within tile == M == N position
    const int hi   = lane >> 4;    // wave half

    const int w0 = (blockIdx.x * 8 + wave) * 16;
    const int h  = blockIdx.y;
    const int b  = blockIdx.z;

    const float* qb = q   + (size_t)b * CHW;
    const float* kb = k   + (size_t)b * CHW;
    const float* vb = v   + (size_t)b * CHW;
    float*       ob = out + (size_t)b * CHW;

    // ---- Preload A: A[M=p, K=c] = q[c, h, w0+p].
    // K-block t covers channels 4t + 2*hi + {0,1} per the 16x4 f32 A layout.
    v2f A[16];
    {
        const float* qp = qb + (size_t)(2 * hi) * HWSZ + (size_t)h * WW + (w0 + p);
        #pragma unroll
        for (int t = 0; t < 16; ++t) {
            A[t].x = qp[(size_t)(4 * t)     * HWSZ];
            A[t].y = qp[(size_t)(4 * t + 1) * HWSZ];
        }
    }

    // q edge columns (channels lane and lane+32) for the two out-of-tile band entries
    const size_t rowq = (size_t)h * WW;
    const float qe0a  = qb[(size_t)lane        * HWSZ + rowq + w0];
    const float qe0b  = qb[(size_t)(lane + 32) * HWSZ + rowq + w0];
    const float qe15a = qb[(size_t)lane        * HWSZ + rowq + (w0 + 15)];
    const float qe15b = qb[(size_t)(lane + 32) * HWSZ + rowq + (w0 + 15)];

    float sc[9];

    #pragma unroll
    for (int dy = 0; dy < 3; ++dy) {
        const int hk = h + dy - 1;
        float s0 = 0.f, s1 = 0.f, s2 = 0.f;   // reference zero-pads k -> OOB score = 0
        if (hk >= 0 && hk < HH) {              // block-uniform: EXEC stays all-1s
            // ---- S[p, n] = sum_c q[c, h, w0+p] * k[c, hk, w0+n] via chained WMMA
            v8f Cm = {};
            const float* kp = kb + (size_t)(2 * hi) * HWSZ + (size_t)hk * WW + (w0 + p);
            #pragma unroll
            for (int t = 0; t < 16; ++t) {
                v2f Bm;
                Bm.x = kp[(size_t)(4 * t)     * HWSZ];
                Bm.y = kp[(size_t)(4 * t + 1) * HWSZ];
                Cm = __builtin_amdgcn_wmma_f32_16x16x4_f32(
                        false, A[t], false, Bm, (short)0, Cm, false, false);
            }
            // scatter C to per-wave LDS: lane<16 holds M=j, lane>=16 holds M=j+8, N=p
            #pragma unroll
            for (int j = 0; j < 8; ++j)
                sS[wave][(j + 8 * hi) * 16 + p] = Cm[j];
            asm volatile("s_wait_dscnt 0x0" ::: "memory");

            // ---- out-of-tile band entries: s(0, w0-1), s(15, w0+16)
            float keL0 = 0.f, keL1 = 0.f, keR0 = 0.f, keR1 = 0.f;
            const size_t rowk = (size_t)hk * WW;
            if (w0 > 0) {
                keL0 = kb[(size_t)lane        * HWSZ + rowk + (w0 - 1)];
                keL1 = kb[(size_t)(lane + 32) * HWSZ + rowk + (w0 - 1)];
            }
            if (w0 + 16 < WW) {
                keR0 = kb[(size_t)lane        * HWSZ + rowk + (w0 + 16)];
                keR1 = kb[(size_t)(lane + 32) * HWSZ + rowk + (w0 + 16)];
            }
            float sL = qe0a * keL0 + qe0b * keL1;
            float sR = qe15a * keR0 + qe15b * keR1;
            #pragma unroll
            for (int off = 16; off > 0; off >>= 1) {
                sL += __shfl_xor(sL, off, 32);
                sR += __shfl_xor(sR, off, 32);
            }

            // ---- gather tridiagonal band for this lane's pixel
            const int cm = (p - 1 < 0) ? 0 : p - 1;
            const int cp = (p + 1 > 15) ? 15 : p + 1;
            s0 = sS[wave][p * 16 + cm];
            s1 = sS[wave][p * 16 + p];
            s2 = sS[wave][p * 16 + cp];
            if (p == 0)  s0 = sL;   // sL==0 when w0==0 (zero-pad)
            if (p == 15) s2 = sR;   // sR==0 when w0+16>=WW
        }
        sc[dy * 3 + 0] = s0;
        sc[dy * 3 + 1] = s1;
        sc[dy * 3 + 2] = s2;
    }

    // ---- softmax over the 9 scores (OOB shifts correctly enter with score 0)
    float mx = sc[0];
    #pragma unroll
    for (int i = 1; i < 9; ++i) mx = fmaxf(mx, sc[i]);
    float wgt[9];
    float sum = 0.f;
    #pragma unroll
    for (int i = 0; i < 9; ++i) { wgt[i] = __expf(sc[i] - mx); sum += wgt[i]; }
    const float inv = 1.0f / sum;
    #pragma unroll
    for (int i = 0; i < 9; ++i) wgt[i] *= inv;

    // ---- weighted sum of shifted v (v zero-padded -> zero weight at W edges)
    const int  wc  = w0 + p;
    const int  wcm = (wc - 1 < 0) ? 0 : wc - 1;
    const int  wcp = (wc + 1 >= WW) ? WW - 1 : wc + 1;
    const bool okL = (wc - 1) >= 0;
    const bool okR = (wc + 1) < WW;

    float wA[3], wB[3], wCx[3];
    #pragma unroll
    for (int dy = 0; dy < 3; ++dy) {
        wA[dy]  = okL ? wgt[dy * 3 + 0] : 0.f;
        wB[dy]  = wgt[dy * 3 + 1];
        wCx[dy] = okR ? wgt[dy * 3 + 2] : 0.f;
    }

    // lanes 0-15 -> channels 0..31, lanes 16-31 -> channels 32..63
    #pragma unroll 4
    for (int c2 = 0; c2 < 32; ++c2) {
        const int c = c2 + 32 * hi;
        const float* vc = vb + (size_t)c * HWSZ;
        float acc = 0.f;
        #pragma unroll
        for (int dy = 0; dy < 3; ++dy) {
            const int hk = h + dy - 1;
            if (hk >= 0 && hk < HH) {   // uniform branch; OOB row contributes 0
                const float* vr = vc + (size_t)hk * WW;
                acc += wA[dy] * vr[wcm] + wB[dy] * vr[wc] + wCx[dy] * vr[wcp];
            }
        }
        ob[(size_t)c * HWSZ + (size_t)h * WW + wc] = acc;
    }
}

extern "C" void kernel_launch(void* const* d_in, const int* in_sizes, int n_in,
                              void* d_out, int out_size, void* d_ws, size_t ws_size,
                              hipStream_t stream) {
    const float* q = (const float*)d_in[0];
    const float* k = (const float*)d_in[1];
    const float* v = (const float*)d_in[2];
    float* out = (float*)d_out;

    const int nb = in_sizes[0] / CHW;      // batch = 8 for the reference shapes
    dim3 grid(WW / (16 * 8), HH, nb);      // (2, 256, 8)
    conv_attn_kernel<<<grid, 256, 0, stream>>>(q, k, v, out);
}